// GRUModel_46986942218988
// MI455X (gfx1250) — compile-verified
//
#include <hip/hip_runtime.h>

// GRU: L=25 layers, E=64, H=8, V=30000, B=1000, S=512.
// Two-kernel design:
//  1) gru_xproj_kernel: bulk-parallel layer-0 input projections via
//     v_wmma_f32_16x16x4_f32, staged in d_ws (24*S*B floats = 49.2 MB).
//  2) gru_recurrent_kernel: 63 persistent 1-wave blocks, one 16-row batch
//     tile each; all weights + hidden state in LDS; 8 fp32 WMMAs per cell.

#define LL 25
#define EE 64
#define HH 8
#define BB 1000
#define SS 512
#define LW 24 /* L-1 */

typedef float v2f __attribute__((ext_vector_type(2)));
typedef float v8f __attribute__((ext_vector_type(8)));

__device__ __forceinline__ v8f wmma4(v2f a, v2f b, v8f c) {
  // D = A(16x4 f32) * B(4x16 f32) + C(16x16 f32)
  return __builtin_amdgcn_wmma_f32_16x16x4_f32(false, a, false, b, (short)0, c,
                                               false, false);
}

__device__ __forceinline__ float sigf(float x) {
  return 1.0f / (1.0f + __expf(-x));
}
__device__ __forceinline__ float tanh_safe(float x) {
  float t = __expf(-2.0f * fabsf(x));      // in (0,1], no overflow
  float r = (1.0f - t) / (1.0f + t);
  return copysignf(r, x);
}

// ---- fragment builders -----------------------------------------------------
// B fragment (4x16): cols 0..7 from Wa[8][8], cols 8..15 from Wb[8][8].
__device__ __forceinline__ v2f bfrag_pair(const float* Wa, const float* Wb,
                                          int c, int lane) {
  int n = lane & 15, nn = n & 7;
  int k0 = 4 * c + ((lane >> 4) << 1);
  const float* W = (n < 8) ? Wa : Wb;
  v2f b;
  b.x = W[(k0 + 0) * HH + nn];
  b.y = W[(k0 + 1) * HH + nn];
  return b;
}
// B fragment: cols 0..7 from Wa, cols 8..15 zero.
__device__ __forceinline__ v2f bfrag_one(const float* Wa, int c, int lane) {
  int n = lane & 15, nn = n & 7;
  int k0 = 4 * c + ((lane >> 4) << 1);
  float w0 = Wa[(k0 + 0) * HH + nn];
  float w1 = Wa[(k0 + 1) * HH + nn];
  v2f b;
  b.x = (n < 8) ? w0 : 0.0f;
  b.y = (n < 8) ? w1 : 0.0f;
  return b;
}
// A fragment (16x4) from LDS tile T[16][8], K chunk c (0..1).
__device__ __forceinline__ v2f afrag(const float* T, int c, int lane) {
  int m = lane & 15;
  int k0 = 4 * c + ((lane >> 4) << 1);
  return *(const v2f*)&T[m * HH + k0];
}
// A fragment of u = h (*) r from two LDS tiles.
__device__ __forceinline__ v2f ufrag(const float* Th, const float* Tr, int c,
                                     int lane) {
  int m = lane & 15;
  int k0 = 4 * c + ((lane >> 4) << 1);
  v2f h2 = *(const v2f*)&Th[m * HH + k0];
  v2f r2 = *(const v2f*)&Tr[m * HH + k0];
  return h2 * r2;
}

// ---- kernel 1: layer-0 input projections ----------------------------------
// xp layout: zr region [(t*16 + n)*B + b], n 0..7 = z, 8..15 = r (bias folded)
//            H  region at offset 16*S*B: [(t*8 + j)*B + b] (bias folded)
__global__ void __launch_bounds__(64)
gru_xproj_kernel(const int* __restrict__ xi, const float* __restrict__ emb,
                 const float* __restrict__ Wxz0, const float* __restrict__ Wxr0,
                 const float* __restrict__ WxH0, const float* __restrict__ bz0,
                 const float* __restrict__ br0, const float* __restrict__ bH0,
                 float* __restrict__ xp) {
  __shared__ float sWz[EE * HH], sWr[EE * HH], sWH[EE * HH], sB[24];
  int tid = threadIdx.x;
  for (int i = tid; i < EE * HH; i += 64) {
    sWz[i] = Wxz0[i];
    sWr[i] = Wxr0[i];
    sWH[i] = WxH0[i];
  }
  if (tid < 8) {
    sB[tid] = bz0[tid];
    sB[8 + tid] = br0[tid];
    sB[16 + tid] = bH0[tid];
  }
  __syncthreads();

  int lane = tid & 31;
  int tile = blockIdx.x * 2 + (tid >> 5);  // 32000 tiles of 16 rows (exact)
  long r0 = (long)tile * 16;
  int n = lane & 15, nn = n & 7;
  int khalf = (lane >> 4) << 1;

  // A-matrix row for this lane (flat row r = t*B + b)
  long rA = r0 + (lane & 15);
  int bA = (int)(rA % BB), tA = (int)(rA / BB);
  int tok = xi[(long)bA * SS + tA];
  const float* erow = emb + (long)tok * EE;

  v8f czr, ch;
#pragma unroll
  for (int v = 0; v < 8; ++v) {
    czr[v] = 0.0f;
    ch[v] = 0.0f;
  }
#pragma unroll
  for (int c = 0; c < 16; ++c) {
    v2f a = *(const v2f*)&erow[4 * c + khalf];
    int k0 = 4 * c + khalf;
    const float* W = (n < 8) ? sWz : sWr;  // packed [Wxz0 | Wxr0]
    v2f bzr;
    bzr.x = W[(k0 + 0) * HH + nn];
    bzr.y = W[(k0 + 1) * HH + nn];
    czr = wmma4(a, bzr, czr);
    float w0 = sWH[(k0 + 0) * HH + nn];
    float w1 = sWH[(k0 + 1) * HH + nn];
    v2f bh;
    bh.x = (n < 8) ? w0 : 0.0f;
    bh.y = (n < 8) ? w1 : 0.0f;
    ch = wmma4(a, bh, ch);
  }
  float bias_zr = (n < 8) ? sB[n] : sB[8 + nn];
  float bias_h = sB[16 + nn];
#pragma unroll
  for (int v = 0; v < 8; ++v) {
    int m = v + ((lane >> 4) << 3);
    long r = r0 + m;
    int b = (int)(r % BB), t = (int)(r / BB);
    xp[((long)t * 16 + n) * BB + b] = czr[v] + bias_zr;
    if (n < 8)
      xp[(long)16 * SS * BB + ((long)t * 8 + n) * BB + b] = ch[v] + bias_h;
  }
}

// ---- kernel 2: recurrence ---------------------------------------------------
__global__ void __launch_bounds__(32)
gru_recurrent_kernel(const float* __restrict__ xp,
                     const float* __restrict__ Whz0,
                     const float* __restrict__ Whr0,
                     const float* __restrict__ WrH0,
                     const float* __restrict__ Wxz,
                     const float* __restrict__ Whz, const float* __restrict__ bz,
                     const float* __restrict__ Wxr,
                     const float* __restrict__ Whr, const float* __restrict__ br,
                     const float* __restrict__ WxH,
                     const float* __restrict__ WrH, const float* __restrict__ bH,
                     const float* __restrict__ Why, const float* __restrict__ by,
                     float* __restrict__ out) {
  __shared__ float sW[LW * 384];   // per layer: Wxz,Whz,Wxr,Whr,WxH,WrH (8x8 each)
  __shared__ float sW0[192];       // Whz0, Whr0, WrH0
  __shared__ float sB[LW * 24];    // per layer: bz, br, bH
  __shared__ float sH[LL * 16 * HH];  // hidden state, all layers, 16 rows
  __shared__ float sR[16 * HH];       // r gate scratch

  int lane = threadIdx.x;
  int tile = blockIdx.x;

  for (int i = lane; i < LW * 64; i += 32) {
    int l = i >> 6, o = i & 63;
    sW[l * 384 + 0 + o]   = Wxz[i];
    sW[l * 384 + 64 + o]  = Whz[i];
    sW[l * 384 + 128 + o] = Wxr[i];
    sW[l * 384 + 192 + o] = Whr[i];
    sW[l * 384 + 256 + o] = WxH[i];
    sW[l * 384 + 320 + o] = WrH[i];
  }
  for (int i = lane; i < 64; i += 32) {
    sW0[i] = Whz0[i];
    sW0[64 + i] = Whr0[i];
    sW0[128 + i] = WrH0[i];
  }
  for (int i = lane; i < LW * 8; i += 32) {
    int l = i >> 3, o = i & 7;
    sB[l * 24 + o] = bz[i];
    sB[l * 24 + 8 + o] = br[i];
    sB[l * 24 + 16 + o] = bH[i];
  }
  for (int i = lane; i < LL * 16 * HH; i += 32) sH[i] = 0.0f;
  __syncthreads();

  int n = lane & 15, nn = n & 7;
  int mhi = (lane >> 4) << 3;  // M offset of this lane's C rows (0 or 8)
  int bidx[8];
#pragma unroll
  for (int v = 0; v < 8; ++v) {
    int b = tile * 16 + (v + mhi);
    bidx[v] = (b < BB) ? b : (BB - 1);  // clamp padded rows (writes guarded)
  }

#pragma unroll 1
  for (int t = 0; t < SS; ++t) {
    // ---------- layer 0 ----------
    float* h0 = &sH[0];
    v8f czr;
#pragma unroll
    for (int v = 0; v < 8; ++v)
      czr[v] = xp[((long)t * 16 + n) * BB + bidx[v]];  // x-proj + bias (z|r)
#pragma unroll
    for (int c = 0; c < 2; ++c)
      czr = wmma4(afrag(h0, c, lane), bfrag_pair(&sW0[0], &sW0[64], c, lane),
                  czr);
    v8f s;
#pragma unroll
    for (int v = 0; v < 8; ++v) s[v] = sigf(czr[v]);  // z (n<8) / r (n>=8)
    if (n >= 8) {
#pragma unroll
      for (int v = 0; v < 8; ++v) sR[(v + mhi) * HH + nn] = s[v];
    }
    __syncthreads();
    v8f ch;
#pragma unroll
    for (int v = 0; v < 8; ++v)
      ch[v] = (n < 8)
                  ? xp[(long)16 * SS * BB + ((long)t * 8 + n) * BB + bidx[v]]
                  : 0.0f;
#pragma unroll
    for (int c = 0; c < 2; ++c)
      ch = wmma4(ufrag(h0, sR, c, lane), bfrag_one(&sW0[128], c, lane), ch);
    __syncthreads();  // all reads of h0 done before overwrite
    if (n < 8) {
#pragma unroll
      for (int v = 0; v < 8; ++v) {
        float hOld = h0[(v + mhi) * HH + n];
        float z = s[v];
        float hc = tanh_safe(ch[v]);
        h0[(v + mhi) * HH + n] = hOld * (1.0f - z) + z * hc;
      }
    }
    __syncthreads();

    // ---------- layers 1..24 ----------
#pragma unroll 1
    for (int l = 0; l < LW; ++l) {
      const float* wl = &sW[l * 384];
      float* hin = &sH[l * 128];        // input = fresh h of layer l
      float* hcur = &sH[(l + 1) * 128]; // this layer's (old) state
      float bzr = (n < 8) ? sB[l * 24 + n] : sB[l * 24 + 8 + nn];
      float bhc = sB[l * 24 + 16 + nn];

      v8f c1;
#pragma unroll
      for (int v = 0; v < 8; ++v) c1[v] = bzr;
#pragma unroll
      for (int c = 0; c < 2; ++c) {
        c1 = wmma4(afrag(hin, c, lane), bfrag_pair(wl + 0, wl + 128, c, lane),
                   c1);  // x@[Wxz|Wxr]
        c1 = wmma4(afrag(hcur, c, lane), bfrag_pair(wl + 64, wl + 192, c, lane),
                   c1);  // h@[Whz|Whr]
      }
      v8f sl;
#pragma unroll
      for (int v = 0; v < 8; ++v) sl[v] = sigf(c1[v]);
      if (n >= 8) {
#pragma unroll
        for (int v = 0; v < 8; ++v) sR[(v + mhi) * HH + nn] = sl[v];
      }
      __syncthreads();

      v8f c2;
#pragma unroll
      for (int v = 0; v < 8; ++v) c2[v] = (n < 8) ? bhc : 0.0f;
#pragma unroll
      for (int c = 0; c < 2; ++c) {
        c2 = wmma4(afrag(hin, c, lane), bfrag_one(wl + 256, c, lane),
                   c2);  // x@WxH
        c2 = wmma4(ufrag(hcur, sR, c, lane), bfrag_one(wl + 320, c, lane),
                   c2);  // (h*r)@WrH
      }
      __syncthreads();  // reads of hcur done before overwrite
      if (n < 8) {
#pragma unroll
        for (int v = 0; v < 8; ++v) {
          float hOld = hcur[(v + mhi) * HH + n];
          float z = sl[v];
          float hc = tanh_safe(c2[v]);
          hcur[(v + mhi) * HH + n] = hOld * (1.0f - z) + z * hc;
        }
      }
      __syncthreads();
    }
  }

  // ---------- outputs ----------
  // logits = h_last[L-1] @ W_hy + b_y   -> out[0..B)
  if (lane < 16) {
    int b = tile * 16 + lane;
    if (b < BB) {
      float acc = by[0];
#pragma unroll
      for (int j = 0; j < HH; ++j)
        acc += sH[(LL - 1) * 128 + lane * HH + j] * Why[j];
      out[b] = acc;
    }
  }
  // h_last [L,B,H] -> out[B ..)
  for (int i = lane; i < LL * 16 * HH; i += 32) {
    int l = i >> 7;
    int rem = i & 127;
    int m = rem >> 3, j = rem & 7;
    int b = tile * 16 + m;
    if (b < BB) out[BB + ((long)l * BB + b) * HH + j] = sH[i];
  }
}

extern "C" void kernel_launch(void* const* d_in, const int* in_sizes, int n_in,
                              void* d_out, int out_size, void* d_ws,
                              size_t ws_size, hipStream_t stream) {
  const int* xi = (const int*)d_in[0];
  const float* emb = (const float*)d_in[1];
  const float* Wxz0 = (const float*)d_in[2];
  const float* Whz0 = (const float*)d_in[3];
  const float* bz0 = (const float*)d_in[4];
  const float* Wxr0 = (const float*)d_in[5];
  const float* Whr0 = (const float*)d_in[6];
  const float* br0 = (const float*)d_in[7];
  const float* WxH0 = (const float*)d_in[8];
  const float* WrH0 = (const float*)d_in[9];
  const float* bH0 = (const float*)d_in[10];
  const float* Wxz = (const float*)d_in[11];
  const float* Whz = (const float*)d_in[12];
  const float* bz = (const float*)d_in[13];
  const float* Wxr = (const float*)d_in[14];
  const float* Whr = (const float*)d_in[15];
  const float* br = (const float*)d_in[16];
  const float* WxH = (const float*)d_in[17];
  const float* WrH = (const float*)d_in[18];
  const float* bH = (const float*)d_in[19];
  const float* Why = (const float*)d_in[20];
  const float* by = (const float*)d_in[21];
  float* xp = (float*)d_ws;  // needs 24*S*B*4 = 49.2 MB scratch
  float* out = (float*)d_out;

  // 32000 16-row tiles, 2 waves per block
  gru_xproj_kernel<<<(SS * BB / 16) / 2, 64, 0, stream>>>(
      xi, emb, Wxz0, Wxr0, WxH0, bz0, br0, bH0, xp);

  int tiles = (BB + 15) / 16;  // 63
  gru_recurrent_kernel<<<tiles, 32, 0, stream>>>(
      xp, Whz0, Whr0, WrH0, Wxz, Whz, bz, Wxr, Whr, br, WxH, WrH, bH, Why, by,
      out);
}